// NeuralRenderer_7988639170906
// MI455X (gfx1250) — compile-verified
//
#include <hip/hip_runtime.h>
#include <hip/hip_bf16.h>

typedef float v2f __attribute__((ext_vector_type(2)));
typedef float v8f __attribute__((ext_vector_type(8)));

#define IMG 256
#define CHUNK 160          // tiles staged in LDS per pass (43.5 KB)
#define SHARP_F (2.0f * (float)IMG)

// ---------------------------------------------------------------------------
// Kernel 1: project vertices (quaternion rotate, scale, translate, y-flip)
// ---------------------------------------------------------------------------
__global__ void proj_kernel(const float* __restrict__ verts,
                            const float* __restrict__ cams,
                            float* __restrict__ vxy, int B, int N) {
    int i = blockIdx.x * blockDim.x + threadIdx.x;
    if (i >= B * N) return;
    int b = i / N;
    const float* c = cams + b * 7;
    float scale = c[0], tx = c[1], ty = c[2];
    float qw = c[3], qx = c[4], qy = c[5], qz = c[6];
    float inv = rsqrtf(qw * qw + qx * qx + qy * qy + qz * qz);
    qw *= inv; qx *= inv; qy *= inv; qz *= inv;
    const float* X = verts + (size_t)i * 3;
    float X0 = X[0], X1 = X[1], X2 = X[2];
    float r00 = 1.f - 2.f * (qy * qy + qz * qz);
    float r01 = 2.f * (qx * qy - qw * qz);
    float r02 = 2.f * (qx * qz + qw * qy);
    float r10 = 2.f * (qx * qy + qw * qz);
    float r11 = 1.f - 2.f * (qx * qx + qz * qz);
    float r12 = 2.f * (qy * qz - qw * qx);
    float vx = scale * (r00 * X0 + r01 * X1 + r02 * X2) + tx;
    float vy = scale * (r10 * X0 + r11 * X1 + r12 * X2) + ty;
    vxy[(size_t)i * 2 + 0] = vx;
    vxy[(size_t)i * 2 + 1] = -vy;   // y-flip folded in
}

// ---------------------------------------------------------------------------
// Kernel 2: per-face edge coefficients, packed in WMMA A-matrix lane layout.
// A tile = 4 faces x 4 edge-rows (4th edge is a +BIG dummy so min ignores it).
// Layout per tile: A[0..31]  = VGPR0 lanes (lane<16: a0(row=lane), else a2(row-16))
//                  A[32..63] = VGPR1 lanes (lane<16: a1(row=lane), else 0)
// Edge value at pixel: e = a0*px + a1*py + a2  (a0=-uy, a1=ux, a2=uy*ax-ux*ay)
// ---------------------------------------------------------------------------
__global__ void facecoef_kernel(const int* __restrict__ faces,
                                const float* __restrict__ vxy,
                                float* __restrict__ gA, float* __restrict__ gS,
                                int B, int N, int F, int ntilesPad) {
    int i = blockIdx.x * blockDim.x + threadIdx.x;
    int facesPad = ntilesPad * 4;
    if (i >= B * facesPad) return;
    int b = i / facesPad;
    int f = i % facesPad;
    int t = f >> 2, fl = f & 3;
    float* A = gA + ((size_t)b * ntilesPad + t) * 64;
    float* S = gS + ((size_t)b * ntilesPad + t) * 4;

    if (f >= F) {   // padded face: force value to -inf so it never wins the max
        for (int e = 0; e < 4; ++e) {
            int r = fl * 4 + e;
            A[r] = 0.f; A[32 + r] = 0.f; A[16 + r] = -3.0e38f; A[48 + r] = 0.f;
        }
        S[fl] = 1.0f;
        return;
    }

    const int* fi = faces + ((size_t)b * F + f) * 3;
    float ax[3], ay[3];
    for (int k = 0; k < 3; ++k) {
        int idx = fi[k];
        const float* v = vxy + ((size_t)b * N + idx) * 2;
        ax[k] = v[0]; ay[k] = v[1];
    }
    float area2 = (ax[1] - ax[0]) * (ay[2] - ay[0]) -
                  (ay[1] - ay[0]) * (ax[2] - ax[0]);
    float s = (area2 > 0.f) ? 1.f : ((area2 < 0.f) ? -1.f : 0.f);
    S[fl] = s;

    for (int e = 0; e < 4; ++e) {
        int r = fl * 4 + e;
        float a0, a1, a2;
        if (e < 3) {
            int ia = e, ib = (e + 1) % 3;
            float ex = ax[ib] - ax[ia], ey = ay[ib] - ay[ia];
            float il = rsqrtf(ex * ex + ey * ey + 1e-12f);
            float ux = ex * il, uy = ey * il;
            a0 = -uy; a1 = ux; a2 = uy * ax[ia] - ux * ay[ia];
        } else {
            a0 = 0.f; a1 = 0.f; a2 = 3.0e38f;  // dummy edge, min ignores
        }
        A[r]      = a0;   // lane r      (V0, K=0 : * px)
        A[32 + r] = a1;   // lane r      (V1, K=1 : * py)
        A[16 + r] = a2;   // lane 16+r   (V0, K=2 : * 1)
        A[48 + r] = 0.f;  // lane 16+r   (V1, K=3 : * 0)
    }
}

// ---------------------------------------------------------------------------
// Kernel 3: render. One wave32 owns 16 pixels of one image row.
// Per tile: 1x v_wmma_f32_16x16x4_f32 evaluates 16 edge-rows (4 faces) at the
// 16 pixels; min over edge-quads, *sign, max-accumulate. Sigmoid applied once.
// ---------------------------------------------------------------------------
__global__ void render_kernel(const float* __restrict__ gA,
                              const float* __restrict__ gS,
                              float* __restrict__ out,
                              int ntilesPad, int nchunks) {
    __shared__ float ldsA[CHUNK * 64];
    __shared__ float ldsS[CHUNK * 4];

    int tid  = threadIdx.x;
    int lane = tid & 31;
    int wave = tid >> 5;
    int lhalf = lane >> 4;          // 0: lanes 0-15, 1: lanes 16-31

    const int blocksPerImg = (IMG * IMG) / 128;   // 128 pixels per block
    int b   = blockIdx.x / blocksPerImg;
    int blk = blockIdx.x % blocksPerImg;
    int pixBase = blk * 128 + wave * 16;
    int y  = pixBase >> 8;          // pixel row
    int x0 = pixBase & (IMG - 1);
    int xpix = x0 + (lane & 15);

    float px = ((float)xpix + 0.5f) * (2.0f / IMG) - 1.0f;
    float py = ((float)y    + 0.5f) * (2.0f / IMG) - 1.0f;

    // B matrix (4x16): K=0 row = px, K=1 row = py, K=2 row = 1, K=3 row = 0
    v2f bm;
    bm.x = lhalf ? 1.0f : px;
    bm.y = lhalf ? 0.0f : py;

    const float* baseA = gA + (size_t)b * ntilesPad * 64;
    const float* baseS = gS + (size_t)b * ntilesPad * 4;

    float mmax = -3.4e38f;

    for (int c = 0; c < nchunks; ++c) {
        __syncthreads();   // protect LDS reuse across chunks
        const float* srcA = baseA + (size_t)c * CHUNK * 64;
        for (int i = tid; i < CHUNK * 64; i += 256) ldsA[i] = srcA[i];
        const float* srcS = baseS + (size_t)c * CHUNK * 4;
        for (int i = tid; i < CHUNK * 4; i += 256) ldsS[i] = srcS[i];
        __syncthreads();

        for (int t = 0; t < CHUNK; ++t) {
            v2f am;
            am.x = ldsA[t * 64 + lane];
            am.y = ldsA[t * 64 + 32 + lane];
            v8f acc = {};
            acc = __builtin_amdgcn_wmma_f32_16x16x4_f32(
                false, am, false, bm, (short)0, acc, false, false);
            // lanes 0-15: VGPR0-3 = face 4t+0, VGPR4-7 = face 4t+1
            // lanes16-31: VGPR0-3 = face 4t+2, VGPR4-7 = face 4t+3
            float mlo = fminf(fminf(acc[0], acc[1]), fminf(acc[2], acc[3]));
            float mhi = fminf(fminf(acc[4], acc[5]), fminf(acc[6], acc[7]));
            float sA = ldsS[t * 4 + 2 * lhalf];
            float sB = ldsS[t * 4 + 2 * lhalf + 1];
            mmax = fmaxf(mmax, fmaxf(sA * mlo, sB * mhi));
        }
    }

    // lanes p and p+16 hold the same pixel over disjoint face subsets:
    // combine with ds_swizzle SWAPX16 (xor 0x10, and 0x1f)
    float other = __int_as_float(
        __builtin_amdgcn_ds_swizzle(__float_as_int(mmax), 0x401f));
    mmax = fmaxf(mmax, other);

    if (lhalf == 0) {
        float val = 1.0f / (1.0f + __expf(-SHARP_F * mmax));
        out[(size_t)b * IMG * IMG + (size_t)y * IMG + xpix] = val;
    }
}

// ---------------------------------------------------------------------------
extern "C" void kernel_launch(void* const* d_in, const int* in_sizes, int n_in,
                              void* d_out, int out_size, void* d_ws, size_t ws_size,
                              hipStream_t stream) {
    const float* verts = (const float*)d_in[0];
    const int*   faces = (const int*)d_in[1];
    const float* cams  = (const float*)d_in[2];
    float* out = (float*)d_out;

    const int B = 2;
    int N = in_sizes[0] / (B * 3);   // 642
    int F = in_sizes[1] / (B * 3);   // 1280
    int ntiles    = (F + 3) / 4;                       // 320
    int nchunks   = (ntiles + CHUNK - 1) / CHUNK;      // 2
    int ntilesPad = nchunks * CHUNK;                   // 320

    float* ws  = (float*)d_ws;
    float* vxy = ws;                                   // B*N*2 floats
    float* gA  = ws + 4096;                            // B*ntilesPad*64
    float* gS  = gA + (size_t)B * ntilesPad * 64;      // B*ntilesPad*4

    int nv = B * N;
    proj_kernel<<<(nv + 255) / 256, 256, 0, stream>>>(verts, cams, vxy, B, N);

    int nf = B * ntilesPad * 4;
    facecoef_kernel<<<(nf + 255) / 256, 256, 0, stream>>>(
        faces, vxy, gA, gS, B, N, F, ntilesPad);

    int nblocks = B * ((IMG * IMG) / 128);             // 1024 blocks x 256 thr
    render_kernel<<<nblocks, 256, 0, stream>>>(gA, gS, out, ntilesPad, nchunks);
}